// ALIGNNConvPyG_54760833024039
// MI455X (gfx1250) — compile-verified
//
#include <hip/hip_runtime.h>
#include <cstddef>

typedef __attribute__((ext_vector_type(2))) float v2f;
typedef __attribute__((ext_vector_type(8))) float v8f;

#define D 128

__device__ __forceinline__ float silu_f(float x) {
    return x / (1.0f + expf(-x));
}

// ---------------------------------------------------------------------------
// proj4: Y_i = X @ W_i + b_i for i=0..3, sharing one A tile.
// Block = 256 threads = 8 waves. Block handles 16 rows; wave w handles cols
// [16w,16w+16). fp32 WMMA 16x16x4, K-loop of 32.
// A frag (16x4 f32): lane<16 -> M=lane, K={k0,k0+1}; lane>=16 -> K={k0+2,k0+3}
// B frag (4x16 f32): lane%16 = N; half selects K pair (k0/k0+2, k0+1/k0+3)
// C (16x16 f32): vgpr r -> M = r + 8*(lane/16), N = lane%16
// ---------------------------------------------------------------------------
__global__ __launch_bounds__(256) void proj4_kernel(
    const float* __restrict__ X,
    const float* __restrict__ W0, const float* __restrict__ B0, float* __restrict__ Y0,
    const float* __restrict__ W1, const float* __restrict__ B1, float* __restrict__ Y1,
    const float* __restrict__ W2, const float* __restrict__ B2, float* __restrict__ Y2,
    const float* __restrict__ W3, const float* __restrict__ B3, float* __restrict__ Y3)
{
    __shared__ __align__(16) float At[16 * D];
    const int m0 = blockIdx.x * 16;
    {   // cooperative A-tile load: 2048 floats = 256 threads x 2 float4
        const float4* s = (const float4*)(X + (size_t)m0 * D);
        float4* d = (float4*)At;
        d[threadIdx.x]       = s[threadIdx.x];
        d[threadIdx.x + 256] = s[threadIdx.x + 256];
    }
    __syncthreads();

    const int wave = threadIdx.x >> 5;
    const int lane = threadIdx.x & 31;
    const int half = lane >> 4;
    const int lm   = lane & 15;
    const int n0   = wave * 16;

    const float* Ws[4] = {W0, W1, W2, W3};
    const float* Bs[4] = {B0, B1, B2, B3};
    float*       Ys[4] = {Y0, Y1, Y2, Y3};

#pragma unroll
    for (int mat = 0; mat < 4; ++mat) {
        const float* W = Ws[mat];
        v8f c = {};
#pragma unroll
        for (int k0 = 0; k0 < D; k0 += 4) {
            const int ka = k0 + 2 * half;
            v2f a, b;
            a.x = At[lm * D + ka];
            a.y = At[lm * D + ka + 1];
            b.x = W[(size_t)ka * D + n0 + lm];
            b.y = W[(size_t)(ka + 1) * D + n0 + lm];
            c = __builtin_amdgcn_wmma_f32_16x16x4_f32(false, a, false, b,
                                                      (short)0, c, false, false);
        }
        const float bias = Bs[mat][n0 + lm];
        float* Y = Ys[mat];
#pragma unroll
        for (int r = 0; r < 8; ++r)
            Y[(size_t)(m0 + r + 8 * half) * D + n0 + lm] = c[r] + bias;
    }
}

// ---------------------------------------------------------------------------
// edge_gate: fuses eg = edge_attr@Weg + beg (WMMA -> LDS) with
// gate = silu(sg[dst] + dg[src] + eg), msg = gate * du[src],
// atomic accumulate into ACC[dst] (pre-initialized with su).
// One block per 16 contiguous edges.
// ---------------------------------------------------------------------------
__global__ __launch_bounds__(256) void edge_gate_kernel(
    const float* __restrict__ EAttr,
    const int*   __restrict__ srcIdx,
    const int*   __restrict__ dstIdx,
    const float* __restrict__ Weg, const float* __restrict__ beg,
    const float* __restrict__ SG,
    const float* __restrict__ DG,
    const float* __restrict__ DU,
    float* __restrict__ ACC)
{
    __shared__ __align__(16) float At[16 * D];
    __shared__ __align__(16) float EG[16 * D];
    __shared__ int sIdx[16];
    __shared__ int dIdx[16];
    const int e0 = blockIdx.x * 16;
    {
        const float4* s = (const float4*)(EAttr + (size_t)e0 * D);
        float4* d = (float4*)At;
        d[threadIdx.x]       = s[threadIdx.x];
        d[threadIdx.x + 256] = s[threadIdx.x + 256];
        if (threadIdx.x < 16) {
            sIdx[threadIdx.x] = srcIdx[e0 + threadIdx.x];
            dIdx[threadIdx.x] = dstIdx[e0 + threadIdx.x];
        }
    }
    __syncthreads();

    const int wave = threadIdx.x >> 5;
    const int lane = threadIdx.x & 31;
    const int half = lane >> 4;
    const int lm   = lane & 15;
    const int n0   = wave * 16;
    {
        v8f c = {};
#pragma unroll
        for (int k0 = 0; k0 < D; k0 += 4) {
            const int ka = k0 + 2 * half;
            v2f a, b;
            a.x = At[lm * D + ka];
            a.y = At[lm * D + ka + 1];
            b.x = Weg[(size_t)ka * D + n0 + lm];
            b.y = Weg[(size_t)(ka + 1) * D + n0 + lm];
            c = __builtin_amdgcn_wmma_f32_16x16x4_f32(false, a, false, b,
                                                      (short)0, c, false, false);
        }
        const float bias = beg[n0 + lm];
#pragma unroll
        for (int r = 0; r < 8; ++r)
            EG[(r + 8 * half) * D + n0 + lm] = c[r] + bias;
    }
    __syncthreads();

    // Phase B: 16 threads per edge, 8 columns per thread (two float4 groups).
    const int el = threadIdx.x >> 4;
    const int cb = (threadIdx.x & 15) * 8;
    const int s  = sIdx[el];
    const int dd = dIdx[el];
    const float* sgRow = SG + (size_t)dd * D + cb;
    const float* dgRow = DG + (size_t)s  * D + cb;
    const float* duRow = DU + (size_t)s  * D + cb;
    const float* egRow = EG + el * D + cb;
    float*       acRow = ACC + (size_t)dd * D + cb;
#pragma unroll
    for (int j = 0; j < 8; j += 4) {
        float4 sv = *(const float4*)(sgRow + j);
        float4 gv = *(const float4*)(dgRow + j);
        float4 uv = *(const float4*)(duRow + j);
        float4 ev = *(const float4*)(egRow + j);
        float g0 = silu_f(sv.x + gv.x + ev.x) * uv.x;
        float g1 = silu_f(sv.y + gv.y + ev.y) * uv.y;
        float g2 = silu_f(sv.z + gv.z + ev.z) * uv.z;
        float g3 = silu_f(sv.w + gv.w + ev.w) * uv.w;
        unsafeAtomicAdd(acRow + j + 0, g0);
        unsafeAtomicAdd(acRow + j + 1, g1);
        unsafeAtomicAdd(acRow + j + 2, g2);
        unsafeAtomicAdd(acRow + j + 3, g3);
    }
}

// ---------------------------------------------------------------------------
// node_out: out = silu(layernorm(ACC)*gamma+beta) + Xin, one wave per row.
// ---------------------------------------------------------------------------
__global__ __launch_bounds__(256) void node_out_kernel(
    const float* __restrict__ ACC,
    const float* __restrict__ Xin,
    const float* __restrict__ gamma, const float* __restrict__ beta,
    float* __restrict__ Out)
{
    const int wave = threadIdx.x >> 5;
    const int lane = threadIdx.x & 31;
    const int row  = blockIdx.x * 8 + wave;
    const int c    = lane * 4;

    float4 h = *(const float4*)(ACC + (size_t)row * D + c);
    float s = h.x + h.y + h.z + h.w;
#pragma unroll
    for (int off = 16; off >= 1; off >>= 1) s += __shfl_xor(s, off, 32);
    const float mean = s * (1.0f / (float)D);

    const float dx = h.x - mean, dy = h.y - mean, dz = h.z - mean, dw = h.w - mean;
    float q = dx * dx + dy * dy + dz * dz + dw * dw;
#pragma unroll
    for (int off = 16; off >= 1; off >>= 1) q += __shfl_xor(q, off, 32);
    const float inv = rsqrtf(q * (1.0f / (float)D) + 1e-5f);

    float4 gm = *(const float4*)(gamma + c);
    float4 bt = *(const float4*)(beta + c);
    float4 xr = *(const float4*)(Xin + (size_t)row * D + c);

    float y0 = dx * inv * gm.x + bt.x;
    float y1 = dy * inv * gm.y + bt.y;
    float y2 = dz * inv * gm.z + bt.z;
    float y3 = dw * inv * gm.w + bt.w;

    float4 o;
    o.x = silu_f(y0) + xr.x;
    o.y = silu_f(y1) + xr.y;
    o.z = silu_f(y2) + xr.z;
    o.w = silu_f(y3) + xr.w;
    *(float4*)(Out + (size_t)row * D + c) = o;
}

extern "C" void kernel_launch(void* const* d_in, const int* in_sizes, int n_in,
                              void* d_out, int out_size, void* d_ws, size_t ws_size,
                              hipStream_t stream) {
    (void)n_in; (void)out_size; (void)ws_size;

    const float* x            = (const float*)d_in[0];
    const float* edge_x       = (const float*)d_in[1];
    const float* lg_edge_attr = (const float*)d_in[2];
    const int*   g_ei         = (const int*)d_in[3];
    const int*   lg_ei        = (const int*)d_in[4];

    const int N   = in_sizes[0] / D;   // 50000
    const int E   = in_sizes[1] / D;   // 500000
    const int ELG = in_sizes[2] / D;   // 500000

    auto F = [&](int i) { return (const float*)d_in[i]; };
    // 5:eu_Wsg 6:eu_bsg 7:eu_Wdg 8:eu_bdg 9:eu_Weg 10:eu_beg
    // 11:eu_Wsu 12:eu_bsu 13:eu_Wdu 14:eu_bdu 15:eu_gamma 16:eu_beta
    // 17..28: nu_* same order

    float* ws  = (float*)d_ws;
    const size_t ED = (size_t)E * D;
    float* sg1  = ws;
    float* dg1  = ws + ED;
    float* du1  = ws + 2 * ED;
    float* acc1 = ws + 3 * ED;

    float* xn_out = (float*)d_out;
    float* ea_out = (float*)d_out + (size_t)N * D;

    // ---------------- Layer 1: edge update on line graph ----------------
    proj4_kernel<<<E / 16, 256, 0, stream>>>(edge_x,
        F(5),  F(6),  sg1,    // Wsg,bsg -> sg (gathered at dst)
        F(7),  F(8),  dg1,    // Wdg,bdg -> dg (gathered at src)
        F(13), F(14), du1,    // Wdu,bdu -> du (gathered at src)
        F(11), F(12), acc1);  // Wsu,bsu -> accumulator init (su)

    edge_gate_kernel<<<ELG / 16, 256, 0, stream>>>(lg_edge_attr,
        lg_ei, lg_ei + ELG,   // src = row0, dst = row1
        F(9), F(10), sg1, dg1, du1, acc1);

    node_out_kernel<<<E / 8, 256, 0, stream>>>(acc1, edge_x, F(15), F(16), ea_out);

    // ---------------- Layer 2: node update on bond graph ----------------
    proj4_kernel<<<N / 16, 256, 0, stream>>>(x,
        F(17), F(18), sg1,
        F(19), F(20), dg1,
        F(25), F(26), du1,
        F(23), F(24), acc1);

    edge_gate_kernel<<<E / 16, 256, 0, stream>>>(ea_out,
        g_ei, g_ei + E,
        F(21), F(22), sg1, dg1, du1, acc1);

    node_out_kernel<<<N / 8, 256, 0, stream>>>(acc1, x, F(27), F(28), xn_out);
}